// Loss4GridSamplingWithFlowVecByFilterFlow_OcclusionAware_85203561218655
// MI455X (gfx1250) — compile-verified
//
#include <hip/hip_runtime.h>
#include <math.h>

// ---------------------------------------------------------------------------
// Loss4GridSamplingWithFlowVecByFilterFlow_OcclusionAware  (gfx1250 / CDNA5)
//
// Bandwidth-bound: filter_flow is 254 MB streamed once (floor ~12us @23.3TB/s).
// K2=121 reduction against the constant offset vectors done with
// V_WMMA_F32_16X16X4_F32 (exact f32): A = [xv;yv] offset rows, B = weights for
// 16 pixels x 4 k-planes, D rows 0/1 = flow_x/flow_y per pixel-lane.
// A-matrix values are generated incrementally (no div/mod, no divergence).
// ---------------------------------------------------------------------------

typedef __attribute__((ext_vector_type(2))) float v2f;
typedef __attribute__((ext_vector_type(8))) float v8f;

namespace cfg {
constexpr int B = 4, C = 3, H = 256, W = 512, K2 = 121;
constexpr int HW = H * W;
constexpr float EPS2 = 1e-3f * 1e-3f;
constexpr int WAVES_PER_BLOCK = 8;             // 256 threads, wave32
constexpr int GROUPS = B * H * (W / 16);       // 32768 pixel-groups of 16
constexpr int NBLOCKS = GROUPS / WAVES_PER_BLOCK;  // 4096
}

__global__ __launch_bounds__(256) void flowloss_main(
    const float* __restrict__ img1, const float* __restrict__ img2,
    const float* __restrict__ ff,   const float* __restrict__ vmask,
    float* __restrict__ partial)
{
  using namespace cfg;
  const int tid  = threadIdx.x;
  const int lane = tid & 31;
  const int wave = tid >> 5;
  const int g    = blockIdx.x * WAVES_PER_BLOCK + wave;   // pixel-group id

  const int wg = W / 16;
  const int w0 = (g % wg) * 16;
  const int h  = (g / wg) % H;
  const int b  =  g / (wg * H);

  // V_WMMA_F32_16X16X4_F32 operand addressing (wave32):
  //   A (16x4): lane -> M = lane&15 ; VGPR v -> K = 2*(lane>>4) + v
  //   B (4x16): lane -> N = lane&15 ; VGPR v -> K = 2*(lane>>4) + v
  const int M    = lane & 15;
  const int kHi  = (lane >> 4) * 2;
  const int pixw = w0 + M;                                // pixel column == N

  const float* wp = ff + (size_t)b * K2 * HW + (size_t)kHi * HW
                       + (size_t)h * W + pixw;

  // Loop-invariant row selectors: only rows M=0 (xv) and M=1 (yv) are nonzero.
  const float selX = (M == 0) ? 1.0f : 0.0f;
  const float selY = (M == 1) ? 1.0f : 0.0f;

  // Incremental offset state for this lane's two K-slots:
  //   slot0: k = chunk*4 + kHi      slot1: k = chunk*4 + kHi + 1
  //   xv[k] = k%11 - 5, yv[k] = k/11 - 5 ; step of 4 < 11 -> at most one wrap.
  float ox0 = (float)kHi - 5.0f, oy0 = -5.0f;        // kHi in {0,2}
  float ox1 = (float)kHi - 4.0f, oy1 = -5.0f;

  v8f acc = {};

  // 30 full chunks of K=4 cover k = 0..119; single-pass coalesced NT stream.
  #pragma unroll 6
  for (int chunk = 0; chunk < 30; ++chunk) {
    v2f bmat;                        // weights: 16 pixels x 4 k-planes
    bmat.x = __builtin_nontemporal_load(wp);
    bmat.y = __builtin_nontemporal_load(wp + HW);
    wp += 4 * (size_t)HW;

    v2f amat;                        // offsets: row0 = xv, row1 = yv
    amat.x = selX * ox0 + selY * oy0;
    amat.y = selX * ox1 + selY * oy1;

    acc = __builtin_amdgcn_wmma_f32_16x16x4_f32(false, amat, false, bmat,
                                                (short)0, acc, false, false);

    // advance both K-slots by 4 (branchless wrap)
    ox0 += 4.0f;
    ox1 += 4.0f;
    const float wr0 = (ox0 > 5.0f) ? 1.0f : 0.0f;
    const float wr1 = (ox1 > 5.0f) ? 1.0f : 0.0f;
    ox0 -= 11.0f * wr0;  oy0 += wr0;
    ox1 -= 11.0f * wr1;  oy1 += wr1;
  }

  // Epilogue chunk: k = 120..123; only k=120 (kHi==0 lanes, slot0) is real.
  {
    const bool k0ok = (120 + kHi) < K2;
    v2f bmat;
    bmat.x = k0ok ? __builtin_nontemporal_load(wp) : 0.0f;
    bmat.y = 0.0f;                   // slot1 k >= 121 for every lane
    v2f amat;
    amat.x = k0ok ? (selX * ox0 + selY * oy0) : 0.0f;  // xv(120)=yv(120)=5
    amat.y = 0.0f;
    acc = __builtin_amdgcn_wmma_f32_16x16x4_f32(false, amat, false, bmat,
                                                (short)0, acc, false, false);
  }

  // D layout: VGPR r, lanes 0-15 hold row M=r, col N=lane.
  const float fx = acc[0];           // flow_x for pixel N=lane   (lanes 0..15)
  const float fy = acc[1];           // flow_y for pixel N=lane   (lanes 0..15)

  float lsum = 0.0f;
  if (lane < 16) {
    const float gx = (float)pixw + fx;
    const float gy = (float)h    + fy;

    const float x0f = floorf(gx), y0f = floorf(gy);
    const float wx1 = gx - x0f, wx0 = 1.0f - wx1;
    const float wy1 = gy - y0f, wy0 = 1.0f - wy1;
    const int x0 = (int)x0f, y0 = (int)y0f;
    const int x1 = x0 + 1,   y1 = y0 + 1;

    const bool v00 = (x0 >= 0) & (x0 < W) & (y0 >= 0) & (y0 < H);
    const bool v01 = (x1 >= 0) & (x1 < W) & (y0 >= 0) & (y0 < H);
    const bool v10 = (x0 >= 0) & (x0 < W) & (y1 >= 0) & (y1 < H);
    const bool v11 = (x1 >= 0) & (x1 < W) & (y1 >= 0) & (y1 < H);

    const float w00 = wx0 * wy0, w01 = wx1 * wy0;
    const float w10 = wx0 * wy1, w11 = wx1 * wy1;

    const float msum = (v00 ? w00 : 0.0f) + (v01 ? w01 : 0.0f)
                     + (v10 ? w10 : 0.0f) + (v11 ? w11 : 0.0f);
    const float maskb = (msum < 0.9999f) ? 0.0f : 1.0f;

    const int xc0 = min(max(x0, 0), W - 1), xc1 = min(max(x1, 0), W - 1);
    const int yc0 = min(max(y0, 0), H - 1), yc1 = min(max(y1, 0), H - 1);
    const int i00 = yc0 * W + xc0, i01 = yc0 * W + xc1;
    const int i10 = yc1 * W + xc0, i11 = yc1 * W + xc1;
    const int ipix = h * W + pixw;

    #pragma unroll
    for (int c = 0; c < C; ++c) {
      const size_t cbase = ((size_t)b * C + c) * HW;
      const float* p1 = img1 + cbase;
      float r = (v00 ? p1[i00] * w00 : 0.0f)
              + (v01 ? p1[i01] * w01 : 0.0f)
              + (v10 ? p1[i10] * w10 : 0.0f)
              + (v11 ? p1[i11] * w11 : 0.0f);
      r *= maskb;
      const float t  = r - img2[cbase + ipix];
      const float vm = vmask[cbase + ipix];
      lsum += vm * maskb * sqrtf(t * t + EPS2);
    }
  }

  // Deterministic block reduction -> one partial per block.
  __shared__ float red[256];
  red[tid] = lsum;
  __syncthreads();
  #pragma unroll
  for (int s = 128; s > 0; s >>= 1) {
    if (tid < s) red[tid] += red[tid + s];
    __syncthreads();
  }
  if (tid == 0) partial[blockIdx.x] = red[0];
}

__global__ __launch_bounds__(256) void flowloss_reduce(
    const float* __restrict__ partial, float* __restrict__ out)
{
  using namespace cfg;
  __shared__ float red[256];
  float s = 0.0f;
  for (int i = threadIdx.x; i < NBLOCKS; i += 256) s += partial[i];
  red[threadIdx.x] = s;
  __syncthreads();
  #pragma unroll
  for (int st = 128; st > 0; st >>= 1) {
    if (threadIdx.x < st) red[threadIdx.x] += red[threadIdx.x + st];
    __syncthreads();
  }
  if (threadIdx.x == 0)
    out[0] = red[0] * (1.0f / (float)((size_t)B * C * H * W));  // WEIGHT = 1
}

extern "C" void kernel_launch(void* const* d_in, const int* in_sizes, int n_in,
                              void* d_out, int out_size, void* d_ws, size_t ws_size,
                              hipStream_t stream) {
  (void)in_sizes; (void)n_in; (void)out_size; (void)ws_size;
  const float* img1  = (const float*)d_in[0];
  const float* img2  = (const float*)d_in[1];
  const float* ff    = (const float*)d_in[2];
  const float* vmask = (const float*)d_in[3];
  float* partial = (float*)d_ws;             // cfg::NBLOCKS floats = 16 KB
  float* out     = (float*)d_out;

  flowloss_main<<<cfg::NBLOCKS, 256, 0, stream>>>(img1, img2, ff, vmask, partial);
  flowloss_reduce<<<1, 256, 0, stream>>>(partial, out);
}